// SGCNet_40020505264386
// MI455X (gfx1250) — compile-verified
//
#include <hip/hip_runtime.h>
#include <hip/hip_bf16.h>
#include <math.h>

#define N_NODES     100000
#define N_EDGES     1600000
#define NUM_FEATS   128
#define HIDDEN      16
#define NUM_CLASSES 40

typedef __attribute__((ext_vector_type(2))) float v2f;
typedef __attribute__((ext_vector_type(8))) float v8f;

// ---------------------------------------------------------------------------
// Kernel 1: y1 = x @ W1   (100000x128 @ 128x16), fp32 WMMA 16x16x4.
// Also writes the same tile into h1acc (self-loop initialization).
// One wave per 16-row M tile; N=16 is exactly one tile; K=128 -> 32 WMMA ops.
// ---------------------------------------------------------------------------
__global__ void __launch_bounds__(256) sgc_gemm1_wmma(
    const float* __restrict__ x, const float* __restrict__ W1,
    float* __restrict__ y1, float* __restrict__ h1acc) {
  int wave = (blockIdx.x * blockDim.x + threadIdx.x) >> 5;   // wave-uniform
  if (wave >= N_NODES / 16) return;                          // uniform exit
  int lane = threadIdx.x & 31;
  int l    = lane & 15;
  int half = lane >> 4;

  int m0 = wave * 16;
  // A fragment source: row (m0+l), cols k + 2*half + {0,1} -> contiguous pair
  const float* arow = x + (size_t)(m0 + l) * NUM_FEATS + 2 * half;

  v8f c = {};
#pragma unroll
  for (int k = 0; k < NUM_FEATS; k += 4) {
    v2f a = *(const v2f*)(arow + k);                         // 8B-aligned
    v2f b;
    b.x = W1[(k + 2 * half + 0) * HIDDEN + l];
    b.y = W1[(k + 2 * half + 1) * HIDDEN + l];
    c = __builtin_amdgcn_wmma_f32_16x16x4_f32(false, a, false, b,
                                              (short)0, c, false, false);
  }
#pragma unroll
  for (int v = 0; v < 8; ++v) {
    size_t idx = (size_t)(m0 + v + 8 * half) * HIDDEN + l;
    y1[idx]    = c[v];
    h1acc[idx] = c[v];   // self-loop term: accumulator starts at y1
  }
}

// ---------------------------------------------------------------------------
// Edge scatter in 16-dim feature space:
//   acc[dst[e]][f] += ew[e] * feat[src[e]][f]
// One thread per (edge, feature); 16 consecutive threads share an edge so
// their atomics hit consecutive addresses (coalesce in L2 atomic units).
// ---------------------------------------------------------------------------
__global__ void __launch_bounds__(256) sgc_scatter16(
    const float* __restrict__ feat, float* __restrict__ acc,
    const int* __restrict__ ei, const float* __restrict__ ew) {
  long long tid = (long long)blockIdx.x * blockDim.x + threadIdx.x;
  if (tid >= (long long)N_EDGES * HIDDEN) return;
  int e = (int)(tid >> 4);
  int f = (int)(tid & 15);
  int s = ei[e];             // edge_index[0][e]
  int d = ei[N_EDGES + e];   // edge_index[1][e]
  float v = ew[e] * feat[(size_t)s * HIDDEN + f];
  atomicAdd(&acc[(size_t)d * HIDDEN + f], v);
}

// Elementwise copy (init g2 accumulator with h1 self-loop term).
__global__ void __launch_bounds__(256) sgc_copy(
    const float* __restrict__ in, float* __restrict__ out, int n) {
  int tid = blockIdx.x * blockDim.x + threadIdx.x;
  if (tid < n) out[tid] = in[tid];
}

// ---------------------------------------------------------------------------
// Kernel 4: out = g2 @ W2   (100000x16 @ 16x40), fp32 WMMA 16x16x4.
// N is tiled 3x16 (cols 40..47 masked). One wave per (Mtile, Ntile).
// ---------------------------------------------------------------------------
__global__ void __launch_bounds__(256) sgc_gemm2_wmma(
    const float* __restrict__ g, const float* __restrict__ W2,
    float* __restrict__ out) {
  const int MT = N_NODES / 16;   // 6250
  const int NT = 3;
  int wave = (blockIdx.x * blockDim.x + threadIdx.x) >> 5;   // wave-uniform
  if (wave >= MT * NT) return;                               // uniform exit
  int lane = threadIdx.x & 31;
  int l    = lane & 15;
  int half = lane >> 4;

  int mt = wave / NT, nt = wave % NT;
  int m0 = mt * 16, col = nt * 16 + l;
  bool colok = (col < NUM_CLASSES);

  v8f c = {};
#pragma unroll
  for (int k = 0; k < HIDDEN; k += 4) {
    v2f a = *(const v2f*)(g + (size_t)(m0 + l) * HIDDEN + k + 2 * half);
    v2f b;
    b.x = colok ? W2[(k + 2 * half + 0) * NUM_CLASSES + col] : 0.0f;
    b.y = colok ? W2[(k + 2 * half + 1) * NUM_CLASSES + col] : 0.0f;
    c = __builtin_amdgcn_wmma_f32_16x16x4_f32(false, a, false, b,
                                              (short)0, c, false, false);
  }
  if (colok) {
#pragma unroll
    for (int v = 0; v < 8; ++v)
      out[(size_t)(m0 + v + 8 * half) * NUM_CLASSES + col] = c[v];
  }
}

// ---------------------------------------------------------------------------
// Kernel 5: in-place log_softmax over rows of 40. One wave per row.
// Lane L holds element L; lanes 0..7 also hold element 32+L.
// ---------------------------------------------------------------------------
__global__ void __launch_bounds__(256) sgc_logsoftmax40(float* __restrict__ h) {
  int wave = (blockIdx.x * blockDim.x + threadIdx.x) >> 5;
  if (wave >= N_NODES) return;
  int lane = threadIdx.x & 31;
  float* row = h + (size_t)wave * NUM_CLASSES;

  float x1 = row[lane];
  float x2 = (lane < 8) ? row[32 + lane] : -INFINITY;

  float m = fmaxf(x1, x2);
#pragma unroll
  for (int off = 16; off > 0; off >>= 1)
    m = fmaxf(m, __shfl_xor(m, off, 32));

  float s = expf(x1 - m) + ((lane < 8) ? expf(x2 - m) : 0.0f);
#pragma unroll
  for (int off = 16; off > 0; off >>= 1)
    s += __shfl_xor(s, off, 32);

  float lz = logf(s) + m;
  row[lane] = x1 - lz;
  if (lane < 8) row[32 + lane] = x2 - lz;
}

// ---------------------------------------------------------------------------
extern "C" void kernel_launch(void* const* d_in, const int* in_sizes, int n_in,
                              void* d_out, int out_size, void* d_ws, size_t ws_size,
                              hipStream_t stream) {
  const float* x  = (const float*)d_in[0];   // [100000,128]
  const float* ew = (const float*)d_in[1];   // [1600000]
  const float* W1 = (const float*)d_in[2];   // [128,16]
  const float* W2 = (const float*)d_in[3];   // [16,40]
  const int*   ei = (const int*)d_in[4];     // [2,1600000]
  float* out = (float*)d_out;                // [100000,40]

  float* y1 = (float*)d_ws;                  // 6.4 MB
  float* h1 = y1 + (size_t)N_NODES * HIDDEN; // 6.4 MB
  float* g2 = h1 + (size_t)N_NODES * HIDDEN; // 6.4 MB

  // 1) y1 = x @ W1 ; h1 = y1 (self loop)
  {
    int waves = N_NODES / 16;                              // 6250
    int blocks = (waves * 32 + 255) / 256;                 // 782
    sgc_gemm1_wmma<<<blocks, 256, 0, stream>>>(x, W1, y1, h1);
  }
  // 2) h1 += scatter_add(ew * y1[src])
  {
    long long total = (long long)N_EDGES * HIDDEN;         // 25.6M
    int blocks = (int)((total + 255) / 256);               // 100000
    sgc_scatter16<<<blocks, 256, 0, stream>>>(y1, h1, ei, ew);
  }
  // 3) g2 = h1 ; g2 += scatter_add(ew * h1[src])
  {
    int n = N_NODES * HIDDEN;
    sgc_copy<<<(n + 255) / 256, 256, 0, stream>>>(h1, g2, n);
    long long total = (long long)N_EDGES * HIDDEN;
    int blocks = (int)((total + 255) / 256);
    sgc_scatter16<<<blocks, 256, 0, stream>>>(h1, g2, ei, ew);
  }
  // 4) out = g2 @ W2
  {
    int waves = (N_NODES / 16) * 3;                        // 18750
    int blocks = (waves * 32 + 255) / 256;                 // 2344
    sgc_gemm2_wmma<<<blocks, 256, 0, stream>>>(g2, W2, out);
  }
  // 5) log_softmax rows, in place on d_out
  {
    int blocks = (N_NODES * 32 + 255) / 256;               // 12500
    sgc_logsoftmax40<<<blocks, 256, 0, stream>>>(out);
  }
}